// GNNLayer_3229815407284
// MI455X (gfx1250) — compile-verified
//
#include <hip/hip_runtime.h>

typedef __attribute__((ext_vector_type(2))) float v2f;
typedef __attribute__((ext_vector_type(8))) float v8f;

#define NN_NODES 100000
#define NN_EDGES 1600000
#define DIM 64
#define TILES (NN_NODES / 16) /* 6250, exact */

// ---------------------------------------------------------------------------
// Kernel 1: zero the accumulator (d_out doubles as aggr buffer)
// ---------------------------------------------------------------------------
__global__ __launch_bounds__(256) void gnn_zero(float4* __restrict__ p, int n4) {
  int i = blockIdx.x * blockDim.x + threadIdx.x;
  if (i < n4) p[i] = make_float4(0.f, 0.f, 0.f, 0.f);
}

// ---------------------------------------------------------------------------
// Kernel 2: edge scatter-add.  16 lanes per edge, each lane handles 4
// consecutive features: one coalesced float4 load of x[col], then 4 native
// global_atomic_add_f32 (L2-resident on MI455X: x + aggr = 51 MB << 192 MB L2)
// ---------------------------------------------------------------------------
__global__ __launch_bounds__(256) void gnn_scatter(const float* __restrict__ x,
                                                   const int* __restrict__ row,
                                                   const int* __restrict__ col,
                                                   float* __restrict__ aggr) {
  int t = blockIdx.x * blockDim.x + threadIdx.x;
  int e = t >> 4;
  int q = t & 15; // which 4-float chunk of the 64-wide row
  if (e >= NN_EDGES) return;
  int r = row[e];
  int c = col[e];
  float4 v = *(const float4*)(x + (size_t)c * DIM + q * 4);
  float* dst = aggr + (size_t)r * DIM + q * 4;
  __hip_atomic_fetch_add(dst + 0, v.x, __ATOMIC_RELAXED, __HIP_MEMORY_SCOPE_AGENT);
  __hip_atomic_fetch_add(dst + 1, v.y, __ATOMIC_RELAXED, __HIP_MEMORY_SCOPE_AGENT);
  __hip_atomic_fetch_add(dst + 2, v.z, __ATOMIC_RELAXED, __HIP_MEMORY_SCOPE_AGENT);
  __hip_atomic_fetch_add(dst + 3, v.w, __ATOMIC_RELAXED, __HIP_MEMORY_SCOPE_AGENT);
}

// ---------------------------------------------------------------------------
// Kernel 3: fp32 WMMA GEMM + bias + ReLU, in place on the aggregate buffer.
// One wave computes a 16-node x 64-feature tile via V_WMMA_F32_16X16X4_F32.
//   A (16x4 f32): lane<16 -> M=lane, a={K,K+1}; lane>=16 -> M=lane-16, a={K+2,K+3}
//   B (4x16 f32): lane<16 -> N=lane, b={K,K+1}; lane>=16 -> N=lane-16, b={K+2,K+3}
//   D (16x16 f32): VGPR v, lane<16 -> M=v,N=lane; lane>=16 -> M=v+8,N=lane-16
// ---------------------------------------------------------------------------
__global__ __launch_bounds__(256) void gnn_gemm_relu(float* __restrict__ inout,
                                                     const float* __restrict__ Wg,
                                                     const float* __restrict__ bias) {
  __shared__ float Wl[DIM * DIM]; // 16 KB, shared by 8 waves
  const int tid = threadIdx.x;
  for (int i = tid; i < (DIM * DIM) / 4; i += 256)
    ((float4*)Wl)[i] = ((const float4*)Wg)[i];
  __syncthreads();

  const int wave = tid >> 5;
  const int lane = tid & 31;
  const int tile = blockIdx.x * 8 + wave;
  if (tile >= TILES) return; // wave-uniform: EXEC stays all-1s for WMMA

  const int m  = lane & 15;        // M for A, N for B/D
  const int kh = (lane >> 4) << 1; // K sub-offset 0 or 2
  const int nodeBase = tile * 16;
  const float* arow = inout + (size_t)(nodeBase + m) * DIM + kh;

  v8f acc[4];
#pragma unroll
  for (int t = 0; t < 4; ++t) {
    float bv = bias[t * 16 + m]; // bias[N] identical across all 8 D-rows
    acc[t] = (v8f){bv, bv, bv, bv, bv, bv, bv, bv};
  }

#pragma unroll
  for (int k0 = 0; k0 < DIM; k0 += 4) {
    v2f a = *(const v2f*)(arow + k0); // A fragment, loaded once per k-step
#pragma unroll
    for (int t = 0; t < 4; ++t) {
      // B = W^T, so B[k][n] = W[n][k]; W row-major [out][in] in LDS
      v2f b = *(const v2f*)(Wl + (t * 16 + m) * DIM + k0 + kh);
      acc[t] = __builtin_amdgcn_wmma_f32_16x16x4_f32(
          /*neg_a=*/false, a, /*neg_b=*/false, b,
          /*c_mod=*/(short)0, acc[t], /*reuse_a=*/false, /*reuse_b=*/false);
    }
  }

  const int mAdd = (lane >> 4) << 3; // lanes 16-31 hold M+8
#pragma unroll
  for (int t = 0; t < 4; ++t) {
#pragma unroll
    for (int v = 0; v < 8; ++v) {
      float val = acc[t][v];
      val = val > 0.f ? val : 0.f; // ReLU
      inout[(size_t)(nodeBase + v + mAdd) * DIM + t * 16 + m] = val;
    }
  }
}

// ---------------------------------------------------------------------------
extern "C" void kernel_launch(void* const* d_in, const int* in_sizes, int n_in,
                              void* d_out, int out_size, void* d_ws, size_t ws_size,
                              hipStream_t stream) {
  const float* x  = (const float*)d_in[0];
  const int*   ei = (const int*)d_in[1];   // [2, N_EDGES]: row then col
  const float* W  = (const float*)d_in[2]; // [64,64] row-major [out][in]
  const float* b  = (const float*)d_in[3]; // [64]
  float* out = (float*)d_out;              // [100000,64]; also the accumulator

  const int* row = ei;
  const int* col = ei + NN_EDGES;

  // 1) zero the accumulator (d_out)
  int n4 = NN_NODES * DIM / 4;
  gnn_zero<<<(n4 + 255) / 256, 256, 0, stream>>>((float4*)out, n4);

  // 2) scatter-add: 16 threads per edge
  long long threads = (long long)NN_EDGES * 16;
  gnn_scatter<<<(unsigned)((threads + 255) / 256), 256, 0, stream>>>(x, row, col, out);

  // 3) in-place fp32 WMMA linear + ReLU: 8 tiles (waves) per block
  int blocks = (TILES + 7) / 8;
  gnn_gemm_relu<<<blocks, 256, 0, stream>>>(out, W, b);
}